// UnpadSeqLen_3848290697286
// MI455X (gfx1250) — compile-verified
//
#include <hip/hip_runtime.h>

// UnpadSeqLen: [B,S,H] fp16 -> [sum(seqlen), H] fp16 row gather.
// Pure bandwidth op (~268 MB traffic -> ~11.5 us floor @ 23.3 TB/s).
//
// Two-phase plan:
//   1) build_src_rows: precompute output-row -> input-row map (<=256KB in d_ws).
//   2) unpad_copy_idx: 4 rows per 256-thread block; uniform s_load of the map,
//      then async global->LDS->global B128 copies (ASYNCcnt-tracked), one
//      s_wait_asynccnt 0 between the load and store phases.
// Fallback (tiny d_ws): self-contained kernel with an inline SALU prefix scan.

#define B_CONST 16
#define S_CONST 4096
#define H_CONST 2048                 // fp16 elements per row
#define ROW_BYTES (H_CONST * 2)      // 4096 bytes per row
#define THREADS 256                  // 256 * 16B = one full row per pass
#define CHUNK 4                      // rows per block (16KB LDS staging)

typedef unsigned int v4u __attribute__((ext_vector_type(4)));
typedef int          v4i __attribute__((ext_vector_type(4)));

#if defined(__gfx1250__) &&                                                   \
    __has_builtin(__builtin_amdgcn_global_load_async_to_lds_b128) &&          \
    __has_builtin(__builtin_amdgcn_global_store_async_from_lds_b128)
#define USE_ASYNC_LDS 1
#else
#define USE_ASYNC_LDS 0
#endif

typedef __attribute__((address_space(1))) v4i gv4i_t;  // global
typedef __attribute__((address_space(3))) v4i lv4i_t;  // LDS

static __device__ __forceinline__ void wait_asynccnt0() {
#if __has_builtin(__builtin_amdgcn_s_wait_asynccnt)
  __builtin_amdgcn_s_wait_asynccnt(0);
#else
  asm volatile("s_wait_asynccnt 0" ::: "memory");
#endif
}

// ---------------------------------------------------------------------------
// Phase 1: srcrow[r] = b*S + t for every packed output row r.
// One branch-free 16-step scan per row; trivially cheap vs the 268MB copy.
// ---------------------------------------------------------------------------
__global__ __launch_bounds__(THREADS) void build_src_rows(
    const int* __restrict__ seqlen, int* __restrict__ srcrow, int nrows) {
  const int r = (int)(blockIdx.x * blockDim.x + threadIdx.x);
  if (r >= nrows) return;
  int b = 0, acc = 0;
#pragma unroll
  for (int i = 0; i < B_CONST; ++i) {
    const int l = seqlen[i];
    const bool take = (acc + l) <= r;
    acc += take ? l : 0;
    b   += take ? 1 : 0;
  }
  srcrow[r] = b * S_CONST + (r - acc);
}

// ---------------------------------------------------------------------------
// Phase 2: lean copy kernel. srcrow reads are block-uniform (s_load_b128 of 4
// consecutive ints); data movement is 8 async B128 ops + 1 asynccnt wait.
// ---------------------------------------------------------------------------
__global__ __launch_bounds__(THREADS) void unpad_copy_idx(
    const _Float16* __restrict__ in,     // [B, S, H]
    const int*      __restrict__ srcrow, // [nrows]
    _Float16*       __restrict__ out,    // [nrows, H]
    int nrows) {
  const int r0   = (int)blockIdx.x * CHUNK;
  const int boff = (int)threadIdx.x * 16;  // 16 bytes (B128) per lane

#if USE_ASYNC_LDS
  __shared__ __align__(16) char stage[CHUNK * ROW_BYTES];
  char* dsts[CHUNK];

#pragma unroll
  for (int i = 0; i < CHUNK; ++i) {
    const int r = r0 + i;
    if (r < nrows) {  // block-uniform guard; EXEC stays all-ones
      const size_t sr = (size_t)srcrow[r];  // uniform scalar load
      const char* src = (const char*)in + sr * (size_t)ROW_BYTES;
      dsts[i] = (char*)out + (size_t)r * (size_t)ROW_BYTES;
      __builtin_amdgcn_global_load_async_to_lds_b128(
          (gv4i_t*)(src + boff), (lv4i_t*)(stage + i * ROW_BYTES + boff),
          /*offset=*/0, /*cpol=*/0);
    }
  }

  wait_asynccnt0();  // all loads landed in LDS (own-lane regions)

#pragma unroll
  for (int i = 0; i < CHUNK; ++i) {
    if (r0 + i < nrows) {
      __builtin_amdgcn_global_store_async_from_lds_b128(
          (gv4i_t*)(dsts[i] + boff), (lv4i_t*)(stage + i * ROW_BYTES + boff),
          /*offset=*/0, /*cpol=*/0);
    }
  }
  // S_ENDPGM performs an implicit wait-idle, draining the async stores.
#else
#pragma unroll
  for (int i = 0; i < CHUNK; ++i) {
    const int r = r0 + i;
    if (r < nrows) {
      const size_t sr = (size_t)srcrow[r];
      const char* src = (const char*)in + sr * (size_t)ROW_BYTES;
      char*       dst = (char*)out + (size_t)r * (size_t)ROW_BYTES;
      v4u v = __builtin_nontemporal_load((const v4u*)(src + boff));
      __builtin_nontemporal_store(v, (v4u*)(dst + boff));
    }
  }
#endif
}

// ---------------------------------------------------------------------------
// Fallback (only if d_ws can't hold the row map): self-contained kernel with
// the inline prefix scan, 4 rows per block.
// ---------------------------------------------------------------------------
__global__ __launch_bounds__(THREADS) void unpad_seqlen_rows(
    const _Float16* __restrict__ in, const int* __restrict__ seqlen,
    _Float16* __restrict__ out, int nrows) {
  const int r0 = (int)blockIdx.x * CHUNK;

  int b = 0, acc = 0;
#pragma unroll
  for (int i = 0; i < B_CONST; ++i) {
    const int l = seqlen[i];
    const bool take = (acc + l) <= r0;
    acc += take ? l : 0;
    b   += take ? 1 : 0;
  }
  int t   = r0 - acc;
  int len = seqlen[b < B_CONST ? b : (B_CONST - 1)];

  const int boff = (int)threadIdx.x * 16;

#if USE_ASYNC_LDS
  __shared__ __align__(16) char stage[CHUNK * ROW_BYTES];
  char* dsts[CHUNK];

#pragma unroll
  for (int i = 0; i < CHUNK; ++i) {
    const int r = r0 + i;
    if (r < nrows) {
      const char* src =
          (const char*)in + ((size_t)b * S_CONST + (size_t)t) * ROW_BYTES;
      dsts[i] = (char*)out + (size_t)r * (size_t)ROW_BYTES;
      __builtin_amdgcn_global_load_async_to_lds_b128(
          (gv4i_t*)(src + boff), (lv4i_t*)(stage + i * ROW_BYTES + boff), 0, 0);
      ++t;
      if (t >= len) {
        ++b; t = 0;
        len = seqlen[b < B_CONST ? b : (B_CONST - 1)];
      }
    }
  }

  wait_asynccnt0();

#pragma unroll
  for (int i = 0; i < CHUNK; ++i) {
    if (r0 + i < nrows) {
      __builtin_amdgcn_global_store_async_from_lds_b128(
          (gv4i_t*)(dsts[i] + boff), (lv4i_t*)(stage + i * ROW_BYTES + boff),
          0, 0);
    }
  }
#else
#pragma unroll
  for (int i = 0; i < CHUNK; ++i) {
    const int r = r0 + i;
    if (r < nrows) {
      const char* src =
          (const char*)in + ((size_t)b * S_CONST + (size_t)t) * ROW_BYTES;
      char* dst = (char*)out + (size_t)r * (size_t)ROW_BYTES;
      v4u v = __builtin_nontemporal_load((const v4u*)(src + boff));
      __builtin_nontemporal_store(v, (v4u*)(dst + boff));
      ++t;
      if (t >= len) {
        ++b; t = 0;
        len = seqlen[b < B_CONST ? b : (B_CONST - 1)];
      }
    }
  }
#endif
}

extern "C" void kernel_launch(void* const* d_in, const int* in_sizes, int n_in,
                              void* d_out, int out_size, void* d_ws,
                              size_t ws_size, hipStream_t stream) {
  (void)in_sizes; (void)n_in;

  const _Float16* in     = (const _Float16*)d_in[0];
  const int*      seqlen = (const int*)d_in[1];
  _Float16*       out    = (_Float16*)d_out;

  const int nrows = out_size / H_CONST;  // sum(seqlen)
  if (nrows <= 0) return;

  const unsigned copy_blocks = (unsigned)((nrows + CHUNK - 1) / CHUNK);
  const size_t   map_bytes   = (size_t)nrows * sizeof(int);

  if (d_ws != nullptr && ws_size >= map_bytes) {
    int* srcrow = (int*)d_ws;
    const unsigned map_blocks = (unsigned)((nrows + THREADS - 1) / THREADS);
    build_src_rows<<<dim3(map_blocks), dim3(THREADS), 0, stream>>>(
        seqlen, srcrow, nrows);
    unpad_copy_idx<<<dim3(copy_blocks), dim3(THREADS), 0, stream>>>(
        in, srcrow, out, nrows);
  } else {
    unpad_seqlen_rows<<<dim3(copy_blocks), dim3(THREADS), 0, stream>>>(
        in, seqlen, out, nrows);
  }
}